// RosaLayer_88605175316975
// MI455X (gfx1250) — compile-verified
//
#include <hip/hip_runtime.h>

#define B_ 4
#define T_ 1024
#define M_ 8
#define K_ 16
#define D_ 256

typedef float v2f __attribute__((ext_vector_type(2)));
typedef float v8f __attribute__((ext_vector_type(8)));
typedef int   v4i __attribute__((vector_size(16)));   // matches builtin param type

#if defined(__AMDGCN__) && __has_builtin(__builtin_amdgcn_global_load_async_to_lds_b128)
#define HAVE_ASYNC_LDS 1
#else
#define HAVE_ASYNC_LDS 0
#endif

// Fused kernel: one workgroup per sequence n = b*M + m (32 blocks, 256 threads).
// LDS (dynamic): W[1024][16] f32 | emb[16][256] f32 | q,v,rstart,d,tau int[1024] | best,hist int[16]
__global__ __launch_bounds__(256)
void rosa_fused(const float* __restrict__ qk, const float* __restrict__ vl,
                const float* __restrict__ emb, float* __restrict__ out)
{
    extern __shared__ char smem_raw[];
    float* Wl   = (float*)smem_raw;            // T_*K_ floats
    float* embl = Wl + T_ * K_;                // K_*D_ floats (64 KB offset -> 16B aligned)
    int* q    = (int*)(embl + K_ * D_);
    int* v    = q    + T_;
    int* rst  = v    + T_;
    int* dd   = rst  + T_;
    int* tau  = dd   + T_;
    int* best = tau  + T_;                     // K_ packed (s+1)<<11 | j ; 0 == empty
    int* hist = best + K_;                     // K_

    const int n   = blockIdx.x;
    const int b   = n / M_;
    const int m   = n % M_;
    const int tid = threadIdx.x;

    // ---- Phase 0: stage emb[m] into LDS (async DMA path on CDNA5); argmax q,v ----
#if HAVE_ASYNC_LDS
    {
        // GLOBAL_LOAD_ASYNC_TO_LDS_B128: memory -> LDS without VGPR round trip,
        // tracked by ASYNCcnt; overlaps with the argmax loads below.
        auto gsrc = (__attribute__((address_space(1))) v4i*)(emb + (size_t)m * K_ * D_);
        auto ldst = (__attribute__((address_space(3))) v4i*)(embl);
        for (int c = tid; c < (K_ * D_) / 4; c += 256)
            __builtin_amdgcn_global_load_async_to_lds_b128(gsrc + c, ldst + c, 0, 0);
    }
#else
    for (int j = tid; j < K_ * D_; j += 256) embl[j] = emb[(size_t)m * K_ * D_ + j];
#endif
    for (int t = tid; t < T_; t += 256) {
        const float* pq = qk + ((size_t)(b * T_ + t) * M_ + m) * K_;
        const float* pv = vl + ((size_t)(b * T_ + t) * M_ + m) * K_;
        float bq = pq[0]; int aq = 0;
        float bv = pv[0]; int av = 0;
        for (int k = 1; k < K_; ++k) {
            float x = pq[k]; if (x > bq) { bq = x; aq = k; }
            float y = pv[k]; if (y > bv) { bv = y; av = k; }
        }
        q[t] = aq; v[t] = av; dd[t] = 0;
    }
#if HAVE_ASYNC_LDS
#if __has_builtin(__builtin_amdgcn_s_wait_asynccnt)
    __builtin_amdgcn_s_wait_asynccnt(0);
#else
    asm volatile("s_wait_asynccnt 0x0" ::: "memory");
#endif
#endif
    __syncthreads();

    // ---- Phase 1: serial incremental DP (d[] is monotone => max-structure valid) ----
    if (tid == 0) {
        for (int a = 0; a < K_; ++a) best[a] = 0;
        int curStart = 0;
        for (int i = 0; i < T_; ++i) {
            const int qi = q[i];
            const bool isnew = (i == 0) || (qi != q[i - 1]);
            if (isnew) {
                // fold previous run [curStart, i) into per-a maxima
                for (int j = curStart; j < i; ++j) {
                    int dl   = (j > 0) ? dd[j - 1] : 0;
                    int cand = ((dl + 2) << 11) | j;     // value = dl+1, packed (val+1,j)
                    int a    = q[j];
                    if (cand > best[a]) best[a] = cand;
                }
                curStart = i;
            }
            rst[i] = curStart;
            if (isnew) {
                // rcf row for this run -> histogram over v-values -> W row (weights)
                int cnt = 0;
                for (int a = 0; a < K_; ++a) hist[a] = 0;
                for (int a = 0; a < K_; ++a) {
                    int pk = best[a];
                    if (pk) { int j = pk & 2047; int tj = rst[j]; hist[v[tj]]++; cnt++; }
                }
                if (cnt > 0) {
                    float inv = 1.0f / (float)cnt;
                    for (int e = 0; e < K_; ++e) Wl[i * K_ + e] = (float)hist[e] * inv;
                } else {
                    int e1 = v[i];
                    for (int e = 0; e < K_; ++e) Wl[i * K_ + e] = (e == e1) ? 1.0f : 0.0f;
                }
            }
            // tau + single d update (+ propagate to candidate best_j+1)
            int pk = best[qi];
            if (pk) {
                int j = pk & 2047;
                int s = (pk >> 11) - 1;
                tau[i] = j;
                dd[j]  = s;
                int p = j + 1;
                if (p < curStart) {
                    int cand = ((s + 2) << 11) | p;      // new value s+1 at position p
                    int a2   = q[p];
                    if (cand > best[a2]) best[a2] = cand;
                }
            } else {
                tau[i] = -1;
            }
        }
    }
    __syncthreads();

    // ---- Phase 2: one-hot W rows for non-start positions ----
    for (int t = tid; t < T_; t += 256) {
        bool isnew = (t == 0) || (q[t] != q[t - 1]);
        if (!isnew) {
            int tj = tau[t];
            int e1 = (tj >= 0) ? v[tj] : -1;
            for (int e = 0; e < K_; ++e) Wl[t * K_ + e] = (e == e1) ? 1.0f : 0.0f;
        }
    }
    __syncthreads();

    // ---- Phase 3: out[n] = W (1024x16) @ emb[m] (16x256) via v_wmma_f32_16x16x4_f32 ----
    const int wave = tid >> 5;
    const int lane = tid & 31;
    const int h    = lane >> 4;   // lane half selects K pair / output row block
    const int r    = lane & 15;
    for (int tile = wave; tile < (T_ / 16) * (D_ / 16); tile += 8) {
        const int t_tile = tile >> 4;
        const int d_tile = tile & 15;
        const int trow   = t_tile * 16 + r;
        const int dcol   = d_tile * 16 + r;
        v8f acc = {0.f, 0.f, 0.f, 0.f, 0.f, 0.f, 0.f, 0.f};
#pragma unroll
        for (int c = 0; c < 4; ++c) {
            const int k0 = 4 * c + 2 * h;
            v2f A;  A.x  = Wl[trow * K_ + k0];       A.y  = Wl[trow * K_ + k0 + 1];
            v2f Bm; Bm.x = embl[k0 * D_ + dcol];     Bm.y = embl[(k0 + 1) * D_ + dcol];
            acc = __builtin_amdgcn_wmma_f32_16x16x4_f32(
                false, A, false, Bm, (short)0, acc, false, false);
        }
#pragma unroll
        for (int rr = 0; rr < 8; ++rr) {
            const int t = t_tile * 16 + rr + 8 * h;
            out[((size_t)(b * T_ + t) * M_ + m) * D_ + dcol] = acc[rr];
        }
    }
}

extern "C" void kernel_launch(void* const* d_in, const int* in_sizes, int n_in,
                              void* d_out, int out_size, void* d_ws, size_t ws_size,
                              hipStream_t stream) {
    (void)in_sizes; (void)n_in; (void)out_size; (void)d_ws; (void)ws_size;
    const float* qk  = (const float*)d_in[0];
    const float* vlg = (const float*)d_in[1];
    const float* emb = (const float*)d_in[2];
    float* out = (float*)d_out;
    const size_t smem = (size_t)(T_ * K_ + K_ * D_) * sizeof(float)
                      + (size_t)(5 * T_ + 2 * K_) * sizeof(int);
    rosa_fused<<<B_ * M_, 256, smem, stream>>>(qk, vlg, emb, out);
}